// GraphEncoderGAT_8675833938498
// MI455X (gfx1250) — compile-verified
//
#include <hip/hip_runtime.h>
#include <hip/hip_bf16.h>
#include <math.h>

typedef float v2f __attribute__((ext_vector_type(2)));
typedef float v8f __attribute__((ext_vector_type(8)));

#define HD   128   // H*D
#define DD   64    // D
#define NHEAD 2

// ---------------- layer-0 linear (K = D_IN = 3, scalar) ----------------
__global__ void k_lin_small(const float* __restrict__ x, const float* __restrict__ W,
                            const float* __restrict__ b, float* __restrict__ y,
                            int N, int K) {
  int gid = blockIdx.x * blockDim.x + threadIdx.x;
  if (gid >= N * HD) return;
  int n = gid >> 7, o = gid & (HD - 1);
  float s = b[o];
  for (int k = 0; k < K; ++k) s += x[n * K + k] * W[o * K + k];
  y[(size_t)n * HD + o] = s;
}

// ---------------- WMMA f32 GEMM: Y[N,128] = X[N,64] @ W[128,64]^T + b ---
// grid.x = M tiles (N/16), block = 256 (8 waves, one 16x16 N-tile each)
__global__ __launch_bounds__(256) void k_gemm_wmma(
    const float* __restrict__ X, const float* __restrict__ W,
    const float* __restrict__ b, float* __restrict__ Y, int N) {
  int wave = threadIdx.x >> 5;           // 0..7 -> n-tile
  int lane = threadIdx.x & 31;
  int half = lane >> 4;                  // 0 or 1
  int l16  = lane & 15;
  int m0 = blockIdx.x * 16;
  int n0 = wave * 16;

  v8f acc = {};
  int am = m0 + l16; if (am > N - 1) am = N - 1;   // clamp keeps EXEC full
  const float* xrow = X + (size_t)am * DD;
  const float* wrow = W + (size_t)(n0 + l16) * DD;

  #pragma unroll
  for (int kk = 0; kk < 16; ++kk) {
    int k0 = kk * 4 + half * 2;
    v2f a, bb;
    a.x  = xrow[k0];     a.y  = xrow[k0 + 1];
    bb.x = wrow[k0];     bb.y = wrow[k0 + 1];
    acc = __builtin_amdgcn_wmma_f32_16x16x4_f32(
        false, a, false, bb, (short)0, acc, false, false);
  }

  #pragma unroll
  for (int r = 0; r < 8; ++r) {
    int m = m0 + r + half * 8;
    if (m < N) Y[(size_t)m * HD + n0 + l16] = acc[r] + b[n0 + l16];
  }
}

// ---------------- edge pass 1: logits + segment max (one wave / edge) ---
__device__ __forceinline__ unsigned f2key(float f) {
  unsigned bits = __float_as_uint(f);
  return (bits & 0x80000000u) ? ~bits : (bits ^ 0x80000000u);
}
__device__ __forceinline__ float key2f(unsigned k) {
  return (k & 0x80000000u) ? __uint_as_float(k ^ 0x80000000u)
                           : __uint_as_float(~k);
}

__global__ void k_edge1(const int* __restrict__ ei, int E, int Etot,
                        const float* __restrict__ xl, const float* __restrict__ xr,
                        const float* __restrict__ att,
                        float* __restrict__ logits, unsigned* __restrict__ maxb) {
  int gid = blockIdx.x * blockDim.x + threadIdx.x;
  int e = gid >> 5;
  if (e >= Etot) return;
  int lane = threadIdx.x & 31;
  int h = lane >> 4, l16 = lane & 15;
  int src = (e < E) ? ei[e]     : (e - E);
  int dst = (e < E) ? ei[E + e] : (e - E);

  float4 a = ((const float4*)(xl + (size_t)src * HD + h * DD))[l16];
  float4 r = ((const float4*)(xr + (size_t)dst * HD + h * DD))[l16];
  float4 w = ((const float4*)(att + h * DD))[l16];
  float t, s = 0.f;
  t = a.x + r.x; s += (t > 0.f ? t : 0.2f * t) * w.x;
  t = a.y + r.y; s += (t > 0.f ? t : 0.2f * t) * w.y;
  t = a.z + r.z; s += (t > 0.f ? t : 0.2f * t) * w.z;
  t = a.w + r.w; s += (t > 0.f ? t : 0.2f * t) * w.w;
  #pragma unroll
  for (int off = 8; off; off >>= 1) s += __shfl_xor(s, off, 16);
  if (l16 == 0) {
    logits[(size_t)e * NHEAD + h] = s;
    atomicMax(&maxb[(size_t)dst * NHEAD + h], f2key(s));
  }
}

// ---------------- edge pass 2: exp weights + weighted scatter-add -------
__global__ void k_edge2(const int* __restrict__ ei, int E, int Etot,
                        const float* __restrict__ xl, const float* __restrict__ logits,
                        const unsigned* __restrict__ maxb,
                        float* __restrict__ denom, float* __restrict__ accum) {
  int gid = blockIdx.x * blockDim.x + threadIdx.x;
  int e = gid >> 5;
  if (e >= Etot) return;
  int lane = threadIdx.x & 31;
  int h = lane >> 4, l16 = lane & 15;
  int src = (e < E) ? ei[e]     : (e - E);
  int dst = (e < E) ? ei[E + e] : (e - E);

  float lg = logits[(size_t)e * NHEAD + h];
  float m  = key2f(maxb[(size_t)dst * NHEAD + h]);
  float wgt = __expf(lg - m);
  if (l16 == 0) atomicAdd(&denom[(size_t)dst * NHEAD + h], wgt);

  float4 a = ((const float4*)(xl + (size_t)src * HD + h * DD))[l16];
  float* o = accum + (size_t)dst * HD + h * DD + l16 * 4;
  atomicAdd(o + 0, wgt * a.x);
  atomicAdd(o + 1, wgt * a.y);
  atomicAdd(o + 2, wgt * a.z);
  atomicAdd(o + 3, wgt * a.w);
}

// ---------------- per-node: normalize, head mean, +bias, ELU ------------
__global__ void k_finalize(const float* __restrict__ accum, const float* __restrict__ denom,
                           const float* __restrict__ cb, float* __restrict__ xbuf, int N) {
  int gid = blockIdx.x * blockDim.x + threadIdx.x;
  if (gid >= N * DD) return;
  int i = gid >> 6, d = gid & (DD - 1);
  float v0 = accum[(size_t)i * HD + d]      / (denom[(size_t)i * NHEAD + 0] + 1e-16f);
  float v1 = accum[(size_t)i * HD + DD + d] / (denom[(size_t)i * NHEAD + 1] + 1e-16f);
  float v = 0.5f * (v0 + v1) + cb[d];
  xbuf[gid] = (v > 0.f) ? v : (__expf(v) - 1.f);
}

// ---------------- global mean pool + final linear -----------------------
__global__ void k_count(const int* __restrict__ batch, float* __restrict__ gcnt, int N) {
  int i = blockIdx.x * blockDim.x + threadIdx.x;
  if (i < N) atomicAdd(&gcnt[batch[i]], 1.f);
}
__global__ void k_pool(const int* __restrict__ batch, const float* __restrict__ xbuf,
                       float* __restrict__ gsum, int N) {
  int gid = blockIdx.x * blockDim.x + threadIdx.x;
  if (gid >= N * DD) return;
  int i = gid >> 6, d = gid & (DD - 1);
  atomicAdd(&gsum[(size_t)batch[i] * DD + d], xbuf[gid]);
}
__global__ void k_final(const float* __restrict__ gsum, const float* __restrict__ gcnt,
                        const float* __restrict__ wlin, const float* __restrict__ blin,
                        float* __restrict__ out, int G, int Z) {
  int gid = blockIdx.x * blockDim.x + threadIdx.x;
  if (gid >= G * Z) return;
  int g = gid / Z, z = gid % Z;
  float inv = 1.f / fmaxf(gcnt[g], 1.f);
  float s = blin[z];
  for (int d = 0; d < DD; ++d) s += (gsum[(size_t)g * DD + d] * inv) * wlin[(size_t)z * DD + d];
  out[gid] = s;
}

extern "C" void kernel_launch(void* const* d_in, const int* in_sizes, int n_in,
                              void* d_out, int out_size, void* d_ws, size_t ws_size,
                              hipStream_t stream) {
  const float* x_in  = (const float*)d_in[0];
  const int*   ei    = (const int*)d_in[1];
  const int*   batch = (const int*)d_in[2];
  const float* wlin  = (const float*)d_in[21];
  const float* blin  = (const float*)d_in[22];

  int N    = in_sizes[2];
  int E    = in_sizes[1] / 2;
  int Etot = E + N;
  int K0   = in_sizes[0] / N;       // D_IN
  int Z    = in_sizes[22];
  int G    = out_size / Z;

  float* ws = (float*)d_ws;
  size_t off = 0;
  auto carve = [&](size_t n) { float* p = ws + off; off += (n + 63) & ~(size_t)63; return p; };
  float*    xl     = carve((size_t)N * HD);
  float*    xr     = carve((size_t)N * HD);
  float*    accum  = carve((size_t)N * HD);
  float*    logits = carve((size_t)Etot * NHEAD);
  unsigned* maxb   = (unsigned*)carve((size_t)N * NHEAD);
  float*    denom  = carve((size_t)N * NHEAD);
  float*    xbuf   = carve((size_t)N * DD);
  float*    gsum   = carve((size_t)G * DD);
  float*    gcnt   = carve((size_t)G);
  (void)ws_size; (void)n_in;

  const int TB = 256;
  int edgeBlocks = (int)(((size_t)Etot * 32 + TB - 1) / TB);

  for (int l = 0; l < 3; ++l) {
    const float* wl  = (const float*)d_in[3 + l * 6 + 0];
    const float* bl  = (const float*)d_in[3 + l * 6 + 1];
    const float* wr  = (const float*)d_in[3 + l * 6 + 2];
    const float* br  = (const float*)d_in[3 + l * 6 + 3];
    const float* att = (const float*)d_in[3 + l * 6 + 4];
    const float* cb  = (const float*)d_in[3 + l * 6 + 5];

    hipMemsetAsync(maxb,  0, (size_t)N * NHEAD * sizeof(unsigned), stream);
    hipMemsetAsync(denom, 0, (size_t)N * NHEAD * sizeof(float), stream);
    hipMemsetAsync(accum, 0, (size_t)N * HD * sizeof(float), stream);

    if (l == 0) {
      int blocks = (N * HD + TB - 1) / TB;
      k_lin_small<<<blocks, TB, 0, stream>>>(x_in, wl, bl, xl, N, K0);
      k_lin_small<<<blocks, TB, 0, stream>>>(x_in, wr, br, xr, N, K0);
    } else {
      int mTiles = (N + 15) / 16;
      k_gemm_wmma<<<mTiles, TB, 0, stream>>>(xbuf, wl, bl, xl, N);
      k_gemm_wmma<<<mTiles, TB, 0, stream>>>(xbuf, wr, br, xr, N);
    }

    k_edge1<<<edgeBlocks, TB, 0, stream>>>(ei, E, Etot, xl, xr, att, logits, maxb);
    k_edge2<<<edgeBlocks, TB, 0, stream>>>(ei, E, Etot, xl, logits, maxb, denom, accum);
    k_finalize<<<(N * DD + TB - 1) / TB, TB, 0, stream>>>(accum, denom, cb, xbuf, N);
  }

  hipMemsetAsync(gsum, 0, (size_t)G * DD * sizeof(float), stream);
  hipMemsetAsync(gcnt, 0, (size_t)G * sizeof(float), stream);
  k_count<<<(N + TB - 1) / TB, TB, 0, stream>>>(batch, gcnt, N);
  k_pool<<<(N * DD + TB - 1) / TB, TB, 0, stream>>>(batch, xbuf, gsum, N);
  k_final<<<(G * Z + TB - 1) / TB, TB, 0, stream>>>(gsum, gcnt, wlin, blin, (float*)d_out, G, Z);
}